// DLRM_Net_25048249270804
// MI455X (gfx1250) — compile-verified
//
#include <hip/hip_runtime.h>
#include <hip/hip_bf16.h>
#include <math.h>

typedef __attribute__((ext_vector_type(2))) float v2f;
typedef __attribute__((ext_vector_type(8))) float v8f;

#define BATCH_N   8192
#define EMB_D     64
#define NTAB      26
#define NROWS_E   100000
#define NFEAT     27          // 1 + NTAB
#define NPAIRS    351         // 27*26/2
#define R_DIM     415
#define R_STRIDE  416         // padded to multiple of 4 for K-loop

// ---------------------------------------------------------------------------
// Generic fp32 WMMA GEMM:  C[M x N] = act( A[M x K] @ W[N x K]^T + bias )
// Block = 128 threads (4 waves). Block tile = 64(M) x 64(N).
// Wave w computes rows [w*16, w*16+16) x 64 cols via 4x v_wmma_f32_16x16x4_f32.
// K staged through LDS in 32-wide chunks; K remainder zero-filled.
// ---------------------------------------------------------------------------
#define KT    32
#define LDS_S 34   // 34 floats/row: 8B-aligned b64 reads, breaks bank conflicts

__global__ __launch_bounds__(128)
void gemm_wmma_f32(const float* __restrict__ A, int lda,
                   const float* __restrict__ W, int ldw,
                   const float* __restrict__ bias,
                   float* __restrict__ C, int ldc,
                   int N, int K, int relu)
{
    __shared__ float As[64 * LDS_S];
    __shared__ float Ws[64 * LDS_S];

    const int tid  = threadIdx.x;
    const int w    = tid >> 5;          // wave 0..3
    const int lane = tid & 31;
    const int r16  = lane & 15;
    const int kh   = lane >> 4;         // 0: K pair {0,1}, 1: K pair {2,3}

    const int rowBlock = blockIdx.y * 64;
    const int colBlock = blockIdx.x * 64;

    v8f acc0 = {}, acc1 = {}, acc2 = {}, acc3 = {};

    for (int kk = 0; kk < K; kk += KT) {
        // cooperative stage: 64 rows x 32 k, coalesced, zero-fill past K
        #pragma unroll
        for (int i = 0; i < 16; ++i) {
            int idx = i * 128 + tid;        // 0..2047
            int r   = idx >> 5;
            int k   = idx & 31;
            int gk  = kk + k;
            float av = 0.f, wv = 0.f;
            if (gk < K) {
                av = A[(size_t)(rowBlock + r) * lda + gk];
                wv = W[(size_t)(colBlock + r) * ldw + gk];
            }
            As[r * LDS_S + k] = av;
            Ws[r * LDS_S + k] = wv;
        }
        __syncthreads();

        #pragma unroll
        for (int k2 = 0; k2 < KT; k2 += 4) {
            // A fragment 16x4 (ISA 7.12.2): lane<16 -> K {k2,k2+1}, lane>=16 -> {k2+2,k2+3}
            v2f a = *(const v2f*)&As[(w * 16 + r16) * LDS_S + k2 + kh * 2];
            v2f b0 = *(const v2f*)&Ws[( 0 + r16) * LDS_S + k2 + kh * 2];
            v2f b1 = *(const v2f*)&Ws[(16 + r16) * LDS_S + k2 + kh * 2];
            v2f b2 = *(const v2f*)&Ws[(32 + r16) * LDS_S + k2 + kh * 2];
            v2f b3 = *(const v2f*)&Ws[(48 + r16) * LDS_S + k2 + kh * 2];
            acc0 = __builtin_amdgcn_wmma_f32_16x16x4_f32(false, a, false, b0, (short)0, acc0, false, false);
            acc1 = __builtin_amdgcn_wmma_f32_16x16x4_f32(false, a, false, b1, (short)0, acc1, false, false);
            acc2 = __builtin_amdgcn_wmma_f32_16x16x4_f32(false, a, false, b2, (short)0, acc2, false, false);
            acc3 = __builtin_amdgcn_wmma_f32_16x16x4_f32(false, a, false, b3, (short)0, acc3, false, false);
        }
        __syncthreads();
    }

    // store: D layout — VGPR v: lanes 0-15 row v, lanes 16-31 row v+8
    v8f accs[4] = {acc0, acc1, acc2, acc3};
    #pragma unroll
    for (int s = 0; s < 4; ++s) {
        int col = colBlock + s * 16 + r16;
        float bv = bias[col];
        #pragma unroll
        for (int v = 0; v < 8; ++v) {
            int row = rowBlock + w * 16 + kh * 8 + v;
            float val = accs[s][v] + bv;
            if (relu) val = fmaxf(val, 0.f);
            C[(size_t)row * ldc + col] = val;
        }
    }
}

// ---------------------------------------------------------------------------
// Fused embedding gather (bag-sum of 2) + pairwise interaction.
// 256 threads = 4 samples/block, 64 threads (one embedding dim each) per sample.
// Writes R[b] = [ x(64) | Zflat(351) | 0 pad ], stride 416.
// ---------------------------------------------------------------------------
__global__ __launch_bounds__(256)
void interact_kernel(const float* __restrict__ x,
                     const int*   __restrict__ lS_i,
                     const float* __restrict__ emb,
                     float* __restrict__ R)
{
    __shared__ float T[4][NFEAT][EMB_D + 1];   // +1 pad breaks bank conflicts

    const int tid = threadIdx.x;
    const int g   = tid >> 6;                  // sample slot 0..3
    const int d   = tid & 63;                  // embedding dim
    const int b   = blockIdx.x * 4 + g;

    float xv = x[(size_t)b * EMB_D + d];
    T[g][0][d] = xv;

    #pragma unroll 2
    for (int t = 0; t < NTAB; ++t) {
        int i0 = lS_i[(size_t)t * (BATCH_N * 2) + 2 * b];
        int i1 = lS_i[(size_t)t * (BATCH_N * 2) + 2 * b + 1];
        const float* e = emb + (size_t)t * NROWS_E * EMB_D;
        T[g][1 + t][d] = e[(size_t)i0 * EMB_D + d] + e[(size_t)i1 * EMB_D + d];
    }
    __syncthreads();

    float* Rrow = R + (size_t)b * R_STRIDE;
    Rrow[d] = xv;
    if (d == 0) Rrow[R_DIM] = 0.f;             // zero the pad column

    for (int p = d; p < NPAIRS; p += 64) {
        // row-major lower-triangle enumeration: (1,0),(2,0),(2,1),(3,0)...
        int i = 1;
        while ((i + 1) * i / 2 <= p) ++i;
        int j = p - i * (i - 1) / 2;
        const float* Ti = T[g][i];
        const float* Tj = T[g][j];
        float s = 0.f;
        #pragma unroll 8
        for (int k = 0; k < EMB_D; ++k) s = fmaf(Ti[k], Tj[k], s);
        Rrow[EMB_D + p] = s;
    }
}

// ---------------------------------------------------------------------------
// Final layer: out[m] = sigmoid( dot(X[m, 0:256], W[0, 0:256]) + b ).
// One wave per row, shuffle reduction.
// ---------------------------------------------------------------------------
__global__ __launch_bounds__(256)
void final_kernel(const float* __restrict__ X,
                  const float* __restrict__ W,
                  const float* __restrict__ bias,
                  float* __restrict__ out)
{
    const int wave = threadIdx.x >> 5;
    const int lane = threadIdx.x & 31;
    const int row  = blockIdx.x * 8 + wave;

    const float* xr = X + (size_t)row * 256;
    float s = 0.f;
    #pragma unroll
    for (int k = 0; k < 256; k += 32) s = fmaf(xr[k + lane], W[k + lane], s);

    #pragma unroll
    for (int off = 16; off >= 1; off >>= 1) s += __shfl_xor(s, off, 32);

    if (lane == 0) out[row] = 1.f / (1.f + __expf(-(s + bias[0])));
}

// ---------------------------------------------------------------------------
extern "C" void kernel_launch(void* const* d_in, const int* in_sizes, int n_in,
                              void* d_out, int out_size, void* d_ws, size_t ws_size,
                              hipStream_t stream)
{
    const float* dense_x = (const float*)d_in[0];
    const int*   lS_i    = (const int*)  d_in[1];
    /* d_in[2] = lS_o (unused) */
    const float* emb_W   = (const float*)d_in[3];
    const float* bot_W1  = (const float*)d_in[4];
    const float* bot_b1  = (const float*)d_in[5];
    const float* bot_W2  = (const float*)d_in[6];
    const float* bot_b2  = (const float*)d_in[7];
    const float* bot_W3  = (const float*)d_in[8];
    const float* bot_b3  = (const float*)d_in[9];
    const float* top_W1  = (const float*)d_in[10];
    const float* top_b1  = (const float*)d_in[11];
    const float* top_W2  = (const float*)d_in[12];
    const float* top_b2  = (const float*)d_in[13];
    const float* top_W3  = (const float*)d_in[14];
    const float* top_b3  = (const float*)d_in[15];
    float* out = (float*)d_out;

    // workspace (floats): buf512 | buf256 | x3 | R   (~40 MB total, reused)
    float* buf512 = (float*)d_ws;
    float* buf256 = buf512 + (size_t)BATCH_N * 512;
    float* x3     = buf256 + (size_t)BATCH_N * 256;
    float* R      = x3     + (size_t)BATCH_N * 64;

    const dim3 blk(128);
    const int MB = BATCH_N / 64;   // 128 row-blocks

    // bottom MLP
    gemm_wmma_f32<<<dim3(512 / 64, MB), blk, 0, stream>>>(dense_x, 13,  bot_W1, 13,  bot_b1, buf512, 512, 512, 13,  1);
    gemm_wmma_f32<<<dim3(256 / 64, MB), blk, 0, stream>>>(buf512, 512,  bot_W2, 512, bot_b2, buf256, 256, 256, 512, 1);
    gemm_wmma_f32<<<dim3( 64 / 64, MB), blk, 0, stream>>>(buf256, 256,  bot_W3, 256, bot_b3, x3,     64,  64,  256, 1);

    // embedding gather + pairwise interaction -> R (8192 x 416, padded)
    interact_kernel<<<BATCH_N / 4, 256, 0, stream>>>(x3, lS_i, emb_W, R);

    // top MLP (K=415 handled by zero-fill remainder against padded stride 416)
    gemm_wmma_f32<<<dim3(512 / 64, MB), blk, 0, stream>>>(R, R_STRIDE, top_W1, R_DIM, top_b1, buf512, 512, 512, R_DIM, 1);
    gemm_wmma_f32<<<dim3(256 / 64, MB), blk, 0, stream>>>(buf512, 512, top_W2, 512,   top_b2, buf256, 256, 256, 512,   1);

    // final 256 -> 1 + sigmoid
    final_kernel<<<BATCH_N / 8, 256, 0, stream>>>(buf256, top_W3, top_b3, out);
}